// RadiusGraph_33036888441073
// MI455X (gfx1250) — compile-verified
//
#include <hip/hip_runtime.h>

typedef float v2f __attribute__((ext_vector_type(2)));
typedef float v8f __attribute__((ext_vector_type(8)));

#define NPB   4096
#define NB    8
#define NTOT  (NPB * NB)          // 32768
#define KNBR  32
#define FDIM  64
#define R2    0.0144f             // 0.12^2 as f32, matches jnp f32 compare

// flat float output offsets (tuple concatenated in return order)
#define OFF_FEAT  0
#define OFF_COORD (NTOT * FDIM)                 // 2097152
#define OFF_ESRC  (OFF_COORD + NTOT * 3)        // 2195456
#define OFF_EDST  (OFF_ESRC + NTOT * KNBR)      // 3244032
#define OFF_DEG   (OFF_EDST + NTOT * KNBR)      // 4292608
#define OFF_BATCH (OFF_DEG + NTOT)              // 4325376

// One block = 256 threads = 8 waves; block owns 128 dst rows of one graph.
// 32 blocks per graph, 256 blocks total.
__global__ __launch_bounds__(256)
void radius_graph_wmma(const float* __restrict__ coord, float* __restrict__ out)
{
    // +4 floats of zero padding so the masked K=3 lane-gather can over-read safely
    __shared__ float sc[NPB * 3 + 4];

    const int tid   = threadIdx.x;
    const int b     = blockIdx.x >> 5;          // graph id
    const int gbase = b * NPB;                  // global node base of graph

    // ---- stage this graph's coords into LDS via CDNA5 async global->LDS DMA ----
    // Generic LDS pointer low 32 bits == LDS byte address (aperture truncation rule).
    {
        const float* gptr   = coord + gbase * 3;            // uniform per block
        const unsigned lds0 = (unsigned)(uintptr_t)&sc[0];
        // 12288 floats = 3072 x b128; 256 threads -> 12 copies each
        for (int t4 = tid; t4 < (NPB * 3) / 4; t4 += 256) {
            unsigned byteoff = (unsigned)t4 * 16u;          // global byte offset
            unsigned ldsaddr = lds0 + byteoff;              // LDS byte address
            asm volatile("global_load_async_to_lds_b128 %0, %1, %2"
                         :
                         : "v"(ldsaddr), "v"(byteoff), "s"(gptr)
                         : "memory");
        }
        asm volatile("s_wait_asynccnt 0x0" ::: "memory");
        if (tid < 4) sc[NPB * 3 + tid] = 0.0f;              // zero the pad
    }
    __syncthreads();

    const int wave = tid >> 5;
    const int lane = tid & 31;
    const int i0   = (blockIdx.x & 31) * 128 + wave * 16;  // local row base of strip
    const int n16  = lane & 15;
    const int hi   = lane >> 4;                 // 0: K=0,1 half / 1: K=2,3 half
    const int c0   = hi * 2;
    const float ymask = (hi == 0) ? 1.0f : 0.0f;   // zero-pad K=3 without branching

    // A matrix: 16x4 f32 (rows i0..i0+15, K = x,y,z,0)
    v2f amat;
    amat.x = sc[(i0 + n16) * 3 + c0];
    amat.y = sc[(i0 + n16) * 3 + c0 + 1] * ymask;

    // row norms |x_i|^2, broadcast the 8 values each lane needs for its D half
    const float rx = sc[(i0 + n16) * 3 + 0];
    const float ry = sc[(i0 + n16) * 3 + 1];
    const float rz = sc[(i0 + n16) * 3 + 2];
    const float ni_full = rx * rx + ry * ry + rz * rz;
    float niv[8];
#pragma unroll
    for (int v = 0; v < 8; ++v)
        niv[v] = __shfl(ni_full, v + 8 * hi, 32);

    // per-row (lanes 0..15) edge emission state
    unsigned cnt = 0;
    const int iloc = i0 + n16;                  // for lanes 0..15: the row they own
    const int self_tile = iloc >> 4;            // tile containing the diagonal bit
    const unsigned self_bit = 1u << (iloc & 15);
    const int bperm_src = lane & 7;             // which ballot this lane's row lives in
    const int shamt = (lane & 8) << 1;          // 0 for rows 0..7, 16 for rows 8..15
    const long long ebase = (long long)(gbase + iloc) * KNBR;
    float* __restrict__ esrc = out + OFF_ESRC;
    float* __restrict__ edst = out + OFF_EDST;
    const float fdst = (float)(gbase + iloc);

    for (int jt = 0; jt < NPB / 16; ++jt) {
        const int j0 = jt * 16;

        // B matrix: 4x16 f32 (cols j0..j0+15); uniform-EXEC masked gather
        v2f bmat;
        bmat.x = sc[(j0 + n16) * 3 + c0];
        bmat.y = sc[(j0 + n16) * 3 + c0 + 1] * ymask;

        // column norms |x_j|^2: partial per half, combine across lane halves
        const float pj = bmat.x * bmat.x + bmat.y * bmat.y;
        const float nj = pj + __shfl_xor(pj, 16, 32);

        // Gram tile via f32 WMMA 16x16x4
        v8f acc = {};
        acc = __builtin_amdgcn_wmma_f32_16x16x4_f32(
            /*neg_a=*/false, amat, /*neg_b=*/false, bmat,
            /*c_mod=*/(short)0, acc, /*reuse_a=*/false, /*reuse_b=*/false);

        // d2 = |xi|^2 + |xj|^2 - 2*Gram; ballot the raw compare (one v_cmp per v),
        // then v_writelane packs the 8 uniform ballots into lanes 0..7 of one VGPR.
        // Self-exclusion (~eye) is handled scalar-side below.
        unsigned packed = 0;
#pragma unroll
        for (int v = 0; v < 8; ++v) {
            const float d2 = niv[v] + nj - 2.0f * acc[v];
            unsigned balv = __builtin_amdgcn_ballot_w32(d2 <= R2);
            asm("v_writelane_b32 %0, %1, %2"
                : "+v"(packed)
                : "s"(balv), "i"(v));
        }
        // route ballot scalars to the lanes that own those rows (uniform EXEC)
        const unsigned rmv = (unsigned)__shfl((int)packed, bperm_src, 32);

        // lanes 0..15: append this tile's hits for row `lane`, ascending j
        if (lane < 16 && cnt < KNBR) {
            unsigned rm = (rmv >> shamt) & 0xFFFFu;
            if (jt == self_tile) rm &= ~self_bit;          // drop diagonal (~eye)
            while (rm && cnt < KNBR) {
                const int bit = __builtin_ctz(rm);
                rm &= rm - 1u;
                esrc[ebase + cnt] = (float)(gbase + j0 + bit);
                edst[ebase + cnt] = fdst;
                ++cnt;
            }
        }
    }

    if (lane < 16) {
        for (unsigned k = cnt; k < KNBR; ++k) {            // invalid slots -> N
            esrc[ebase + k] = (float)NTOT;
            edst[ebase + k] = (float)NTOT;
        }
        out[OFF_DEG + gbase + iloc] = (float)cnt;          // degree = min(count, K)
    }
}

// passthrough: features, coords, batch(int->float)
__global__ __launch_bounds__(256)
void passthrough_copy(const float* __restrict__ feat,
                      const float* __restrict__ coord,
                      const int* __restrict__ batch,
                      float* __restrict__ out)
{
    const int total = NTOT * FDIM + NTOT * 3 + NTOT;
    for (int t = blockIdx.x * blockDim.x + threadIdx.x; t < total;
         t += gridDim.x * blockDim.x) {
        if (t < NTOT * FDIM) {
            out[OFF_FEAT + t] = feat[t];
        } else if (t < NTOT * FDIM + NTOT * 3) {
            const int u = t - NTOT * FDIM;
            out[OFF_COORD + u] = coord[u];
        } else {
            const int u = t - NTOT * FDIM - NTOT * 3;
            out[OFF_BATCH + u] = (float)batch[u];
        }
    }
}

extern "C" void kernel_launch(void* const* d_in, const int* in_sizes, int n_in,
                              void* d_out, int out_size, void* d_ws, size_t ws_size,
                              hipStream_t stream) {
    (void)in_sizes; (void)n_in; (void)out_size; (void)d_ws; (void)ws_size;
    const float* coord = (const float*)d_in[0];   // [N,3]
    const float* feat  = (const float*)d_in[1];   // [N,64]
    const int*   batch = (const int*)d_in[2];     // [N]
    float* out = (float*)d_out;

    radius_graph_wmma<<<256, 256, 0, stream>>>(coord, out);

    const int total = NTOT * FDIM + NTOT * 3 + NTOT;
    passthrough_copy<<<(total + 255) / 256, 256, 0, stream>>>(feat, coord, batch, out);
}